// Subtractor12Bit_76312978916084
// MI455X (gfx1250) — compile-verified
//
#include <hip/hip_runtime.h>
#include <stdint.h>

// 12-bit ripple-borrow subtractor on {0,1} spike floats.
// diff bits = bits of (A_int - B_int), borrow = (A_int < B_int).
// Memory-bound streaming kernel: async global->LDS double-buffered pipeline
// (CDNA5 GLOBAL_LOAD_ASYNC_TO_LDS_B128, ASYNCcnt), NT 128-bit stores.

typedef float    f4 __attribute__((ext_vector_type(4)));
typedef uint32_t u4 __attribute__((ext_vector_type(4)));

#define DEVFN static __device__ __forceinline__

// Pack one 12-element row (index 0 = MSB ... index 11 = LSB) into an int.
// Inputs are exactly 0.0f (0x0) or 1.0f (0x3F800000), so (bits != 0) is the bit.
DEVFN uint32_t pack12(u4 v0, u4 v1, u4 v2) {
    uint32_t r = 0;
    r |= (v0.x ? 1u : 0u) << 11;
    r |= (v0.y ? 1u : 0u) << 10;
    r |= (v0.z ? 1u : 0u) << 9;
    r |= (v0.w ? 1u : 0u) << 8;
    r |= (v1.x ? 1u : 0u) << 7;
    r |= (v1.y ? 1u : 0u) << 6;
    r |= (v1.z ? 1u : 0u) << 5;
    r |= (v1.w ? 1u : 0u) << 4;
    r |= (v2.x ? 1u : 0u) << 3;
    r |= (v2.y ? 1u : 0u) << 2;
    r |= (v2.z ? 1u : 0u) << 1;
    r |= (v2.w ? 1u : 0u);
    return r;
}

DEVFN float bitf(uint32_t d, int k) { return ((d >> k) & 1u) ? 1.0f : 0.0f; }

DEVFN void computeStore(uint32_t ai, uint32_t bi,
                        float* __restrict__ diffRow, float* __restrict__ borrowOut) {
    uint32_t d = ai - bi;  // two's complement; bits [11:0] are the 12-bit difference
    f4 o0 = { bitf(d, 11), bitf(d, 10), bitf(d, 9), bitf(d, 8) };
    f4 o1 = { bitf(d, 7),  bitf(d, 6),  bitf(d, 5), bitf(d, 4) };
    f4 o2 = { bitf(d, 3),  bitf(d, 2),  bitf(d, 1), bitf(d, 0) };
    __builtin_nontemporal_store(o0, (f4*)diffRow + 0);
    __builtin_nontemporal_store(o1, (f4*)diffRow + 1);
    __builtin_nontemporal_store(o2, (f4*)diffRow + 2);
    __builtin_nontemporal_store((ai < bi) ? 1.0f : 0.0f, borrowOut);
}

// Async-copy one 48-byte row (3 x b128) from global to this lane's LDS slot.
// INST_OFFSET applies to both the LDS and global address (ISA 08 §4.4).
DEVFN void asyncLoadRow(const float* gsrc, uint32_t ldsByteAddr) {
    uint64_t ga = (uint64_t)(uintptr_t)gsrc;
    asm volatile("global_load_async_to_lds_b128 %0, %1, off"
                 :: "v"(ldsByteAddr), "v"(ga) : "memory");
    asm volatile("global_load_async_to_lds_b128 %0, %1, off offset:16"
                 :: "v"(ldsByteAddr), "v"(ga) : "memory");
    asm volatile("global_load_async_to_lds_b128 %0, %1, off offset:32"
                 :: "v"(ldsByteAddr), "v"(ga) : "memory");
}

#define TILE_ROWS   256
#define ROW_BYTES   48                      // 12 floats
#define HALF_BYTES  (TILE_ROWS * ROW_BYTES) // 12288 bytes (A or B region)
#define BUF_BYTES   (2 * HALF_BYTES)        // 24576 bytes per stage
#define SMEM_BYTES  (2 * BUF_BYTES)         // 49152 bytes, double buffered

__global__ __launch_bounds__(256)
void Subtractor12Bit_76312978916084_kernel(const float* __restrict__ A,
                                           const float* __restrict__ B,
                                           float* __restrict__ out,
                                           int nrows) {
    __shared__ __align__(16) unsigned char smem[SMEM_BYTES];
    const int tid    = (int)threadIdx.x;
    const int ntiles = nrows >> 8;                 // full 256-row tiles
    const int stride = (int)gridDim.x;

    // Wave-relative LDS byte address of smem (low 32 bits of the flat pointer).
    const uint32_t smemBase = (uint32_t)(uintptr_t)(&smem[0]);

    float* __restrict__ diffOut   = out;
    float* __restrict__ borrowOut = out + (size_t)nrows * 12;

    auto issue = [&](int tile, int stage) {
        const int row = (tile << 8) + tid;
        const uint32_t aoff = smemBase + (uint32_t)stage * BUF_BYTES + (uint32_t)tid * ROW_BYTES;
        asyncLoadRow(A + (size_t)row * 12, aoff);               // 3 async b128
        asyncLoadRow(B + (size_t)row * 12, aoff + HALF_BYTES);  // 3 async b128
    };

    int t = (int)blockIdx.x;
    if (t < ntiles) issue(t, 0);   // prologue prefetch
    int stage = 0;

    for (int cur = t; cur < ntiles; cur += stride) {
        const int nxt = cur + stride;
        if (nxt < ntiles) {
            // Previous consumer of the alternate buffer has fully drained its
            // ds reads (data was consumed last iteration); make it explicit.
            asm volatile("s_wait_dscnt 0" ::: "memory");
            issue(nxt, stage ^ 1);
            // 6 newer async ops outstanding -> current tile's 6 are complete.
            asm volatile("s_wait_asynccnt 6" ::: "memory");
        } else {
            asm volatile("s_wait_asynccnt 0" ::: "memory");
        }

        const unsigned char* bufA = smem + stage * BUF_BYTES + tid * ROW_BYTES;
        const u4* pa = (const u4*)bufA;                 // ds_load_b128 x3
        const u4* pb = (const u4*)(bufA + HALF_BYTES);  // ds_load_b128 x3
        const uint32_t ai = pack12(pa[0], pa[1], pa[2]);
        const uint32_t bi = pack12(pb[0], pb[1], pb[2]);

        const int row = (cur << 8) + tid;
        computeStore(ai, bi, diffOut + (size_t)row * 12, borrowOut + row);
        stage ^= 1;
    }

    // Tail rows (nrows not a multiple of 256) — direct path, block 0 only.
    const int tailStart = ntiles << 8;
    const int tail = nrows - tailStart;
    if (blockIdx.x == 0 && tail > 0) {
        for (int i = tid; i < tail; i += (int)blockDim.x) {
            const int row = tailStart + i;
            const u4* ga = (const u4*)(A + (size_t)row * 12);
            const u4* gb = (const u4*)(B + (size_t)row * 12);
            const uint32_t ai = pack12(ga[0], ga[1], ga[2]);
            const uint32_t bi = pack12(gb[0], gb[1], gb[2]);
            computeStore(ai, bi, diffOut + (size_t)row * 12, borrowOut + row);
        }
    }
}

extern "C" void kernel_launch(void* const* d_in, const int* in_sizes, int n_in,
                              void* d_out, int out_size, void* d_ws, size_t ws_size,
                              hipStream_t stream) {
    (void)n_in; (void)out_size; (void)d_ws; (void)ws_size;
    const float* A = (const float*)d_in[0];
    const float* B = (const float*)d_in[1];
    float* out = (float*)d_out;

    const int nrows  = in_sizes[0] / 12;   // BATCH = 4,194,304
    const int ntiles = nrows >> 8;         // 16384 full tiles

    int blocks = 2048;                     // persistent blocks, ~8 tiles each
    if (ntiles > 0 && ntiles < blocks) blocks = ntiles;
    if (blocks < 1) blocks = 1;

    Subtractor12Bit_76312978916084_kernel<<<blocks, 256, 0, stream>>>(A, B, out, nrows);
}